// ScaledDotProductAttention_10101763080208
// MI455X (gfx1250) — compile-verified
//
#include <hip/hip_runtime.h>
#include <hip/hip_bf16.h>
#include <math.h>

// ---------------------------------------------------------------------------
// Reference:
//   scores = q @ k^T / T            (8,4096,4096)  -> only row-max survives
//   s_i    = exp(rowmax_i)          (8,4096)
//   a      = softmax(s, axis=1)     (8,4096)
//   delta  = 16th-largest(a) + eps ; w = max(a - delta, 0); sp = w/(sum w + eps)
//   out[b,i,:] = sp[b,i] * v[b,0,:] ; outputs = (out, sp[:,:,None], 0.0)
// ---------------------------------------------------------------------------

typedef __attribute__((ext_vector_type(2))) float v2f;
typedef __attribute__((ext_vector_type(4))) float v4f;
typedef __attribute__((ext_vector_type(8))) float v8f;

#define TEMP_INV (1.0f / 11.313708498984761f)
#define SEQ 4096
#define HD  128
#define NB  8
#define EPSV 1e-7f

// -------------------------------------------------------------------------
// Kernel A: row-max of (q @ k^T) via V_WMMA_F32_16X16X4_F32.
// Block = 256 threads = 8 waves; each wave owns one 16-row q tile.
// Block stages 16x128 k tiles into LDS (padded stride, conflict-free).
//
// fp32 WMMA fragment layout (ISA 7.12.2, 32-bit A 16x4):
//   lanes 0-15 :  M=lane, VGPR0=K0, VGPR1=K1
//   lanes 16-31:  M=lane-16, VGPR0=K2, VGPR1=K3
// => per lane, A fragment for K-chunk c is float2 at q[row][4c + 2*half].
// B (4x16) mirrors with N across lanes; C/D: VGPR r = row r (lanes 0-15),
// row r+8 (lanes 16-31), N = lane%16.
// -------------------------------------------------------------------------
__global__ __launch_bounds__(256) void rowmax_wmma_kernel(
    const float* __restrict__ q, const float* __restrict__ k,
    float* __restrict__ mOut)
{
  constexpr int LDK = 132;                 // padded row stride (floats)
  __shared__ float kt[16 * LDK];           // one 16x128 k tile

  const int b    = blockIdx.y;
  const int wave = threadIdx.x >> 5;       // 0..7
  const int lane = threadIdx.x & 31;
  const int half = lane >> 4;              // 0 or 1
  const int l    = lane & 15;

  const int qrow0 = blockIdx.x * 128 + wave * 16;
  const float* qb = q + ((size_t)b * SEQ + qrow0) * HD;
  const float* kb = k + (size_t)b * SEQ * HD;

  // Preload the wave's A fragments for all 32 K-chunks (64 VGPRs).
  v2f aF[32];
#pragma unroll
  for (int c = 0; c < 32; ++c)
    aF[c] = *(const v2f*)(qb + (size_t)l * HD + 4 * c + 2 * half);

  float rm[8];
#pragma unroll
  for (int r = 0; r < 8; ++r) rm[r] = -__builtin_inff();

  for (int j = 0; j < SEQ / 16; ++j) {
    __syncthreads();
    // Stage k tile: 16 rows x 128 floats; each thread moves 8 floats.
    {
      const int row = threadIdx.x >> 4;          // 0..15
      const int d0  = (threadIdx.x & 15) * 8;    // 0..120
      const float* src = kb + (size_t)(j * 16 + row) * HD + d0;
      v4f x0 = *(const v4f*)(src);
      v4f x1 = *(const v4f*)(src + 4);
      *(v4f*)&kt[row * LDK + d0]     = x0;
      *(v4f*)&kt[row * LDK + d0 + 4] = x1;
    }
    __syncthreads();

    v8f acc = {};
#pragma unroll
    for (int c = 0; c < 32; ++c) {
      v2f bF = *(const v2f*)&kt[l * LDK + 4 * c + 2 * half];
      // (neg_a, A, neg_b, B, c_mod, C, reuse_a, reuse_b)
      acc = __builtin_amdgcn_wmma_f32_16x16x4_f32(
          false, aF[c], false, bF, (short)0, acc, false, false);
    }
#pragma unroll
    for (int r = 0; r < 8; ++r) rm[r] = fmaxf(rm[r], acc[r]);
  }

  // Row-max: reduce across the 16 lanes of each half-wave.
#pragma unroll
  for (int r = 0; r < 8; ++r) {
    float x = rm[r];
#pragma unroll
    for (int s = 8; s >= 1; s >>= 1)
      x = fmaxf(x, __shfl_xor(x, s, 32));
    rm[r] = x;
  }
  if (l == 0) {
#pragma unroll
    for (int r = 0; r < 8; ++r)
      mOut[(size_t)b * SEQ + qrow0 + half * 8 + r] = rm[r] * TEMP_INV;
  }
}

// -------------------------------------------------------------------------
// Kernel B: per batch -> softmax over 4096, top-16 threshold, renorm,
// write out / attn3 / extra_loss.  One block (256 threads) per batch.
// -------------------------------------------------------------------------
__global__ __launch_bounds__(256) void softmax_topk_out_kernel(
    const float* __restrict__ mIn, const float* __restrict__ v,
    float* __restrict__ dout)
{
  constexpr int PT = SEQ / 256;            // 16 elements per thread
  __shared__ float Awork[SEQ];             // masked working copy / later sp
  __shared__ float Aorig[SEQ];             // original softmax values
  __shared__ float red[256];
  __shared__ int   redi[256];
  __shared__ float vb[HD];
  __shared__ float kthval;

  const int b = blockIdx.x;
  const int t = threadIdx.x;

  // s_i = exp(rowmax_i / T)
  float s[PT];
  float lmax = -__builtin_inff();
#pragma unroll
  for (int u = 0; u < PT; ++u) {
    float mi = mIn[(size_t)b * SEQ + t * PT + u];
    s[u] = __expf(mi);
    lmax = fmaxf(lmax, s[u]);
  }
  red[t] = lmax;
  __syncthreads();
  for (int w = 128; w > 0; w >>= 1) {
    if (t < w) red[t] = fmaxf(red[t], red[t + w]);
    __syncthreads();
  }
  const float smax = red[0];
  __syncthreads();

  // sum of exp(s - smax)
  float e[PT];
  float lsum = 0.f;
#pragma unroll
  for (int u = 0; u < PT; ++u) { e[u] = __expf(s[u] - smax); lsum += e[u]; }
  red[t] = lsum;
  __syncthreads();
  for (int w = 128; w > 0; w >>= 1) {
    if (t < w) red[t] += red[t + w];
    __syncthreads();
  }
  const float ssum = red[0];
  __syncthreads();

#pragma unroll
  for (int u = 0; u < PT; ++u) {
    float a = e[u] / ssum;
    Awork[t * PT + u] = a;
    Aorig[t * PT + u] = a;
  }
  __syncthreads();

  // top-16 via 16 masked argmax passes (value of the 16th is all we need)
  for (int it = 0; it < 16; ++it) {
    float bv = -__builtin_inff();
    int   bi = 0;
#pragma unroll
    for (int u = 0; u < PT; ++u) {
      float x = Awork[t * PT + u];
      if (x > bv) { bv = x; bi = t * PT + u; }
    }
    red[t] = bv; redi[t] = bi;
    __syncthreads();
    for (int w = 128; w > 0; w >>= 1) {
      if (t < w && red[t + w] > red[t]) { red[t] = red[t + w]; redi[t] = redi[t + w]; }
      __syncthreads();
    }
    if (t == 0) {
      if (it == 15) kthval = red[0];
      Awork[redi[0]] = -__builtin_inff();
    }
    __syncthreads();
  }
  const float delta = kthval + EPSV;

  // w = max(a - delta, 0); sp = w / (sum w + eps)
  float wv[PT];
  float lw = 0.f;
#pragma unroll
  for (int u = 0; u < PT; ++u) {
    wv[u] = fmaxf(Aorig[t * PT + u] - delta, 0.f);
    lw += wv[u];
  }
  red[t] = lw;
  __syncthreads();
  for (int w = 128; w > 0; w >>= 1) {
    if (t < w) red[t] += red[t + w];
    __syncthreads();
  }
  const float inv = 1.f / (red[0] + EPSV);
  __syncthreads();

  // store sp into Awork (reuse), write attn3 slice of d_out
  float* out_o = dout;                                  // (8,4096,128)
  float* out_a = dout + (size_t)NB * SEQ * HD;          // (8,4096,1)
#pragma unroll
  for (int u = 0; u < PT; ++u) {
    float sp = wv[u] * inv;
    Awork[t * PT + u] = sp;
    out_a[(size_t)b * SEQ + t * PT + u] = sp;
  }
  if (t < HD) vb[t] = v[(size_t)b * HD + t];
  __syncthreads();

  // out[b,i,:] = sp_i * v[b,0,:]  — coalesced float4 stores
  const int d0 = (t & 31) * 4;
  v4f vv = *(const v4f*)&vb[d0];
  for (int base = 0; base < SEQ; base += 8) {
    int row = base + (t >> 5);
    float sp = Awork[row];
    v4f o = vv * sp;
    *(v4f*)&out_o[((size_t)b * SEQ + row) * HD + d0] = o;
  }

  if (b == 0 && t == 0)
    dout[(size_t)NB * SEQ * HD + (size_t)NB * SEQ] = 0.0f;  // extra_loss
}

// -------------------------------------------------------------------------
extern "C" void kernel_launch(void* const* d_in, const int* in_sizes, int n_in,
                              void* d_out, int out_size, void* d_ws, size_t ws_size,
                              hipStream_t stream) {
  const float* q = (const float*)d_in[0];   // (8,4096,128)
  const float* k = (const float*)d_in[1];   // (8,4096,128)
  const float* v = (const float*)d_in[2];   // (8,1,128)
  float* out = (float*)d_out;
  float* m   = (float*)d_ws;                // 8*4096 floats = 128 KB scratch

  dim3 gridA(SEQ / 128, NB);                // 32 x 8 blocks, 8 waves each
  rowmax_wmma_kernel<<<gridA, 256, 0, stream>>>(q, k, m);
  softmax_topk_out_kernel<<<NB, 256, 0, stream>>>(m, v, out);
}